// BayesLinearMF_20950850470570
// MI455X (gfx1250) — compile-verified
//
#include <hip/hip_runtime.h>

#define BS  256
#define OUT 1024
#define IN  1024

typedef float v2f __attribute__((ext_vector_type(2)));
typedef float v4f __attribute__((ext_vector_type(4)));
typedef float v8f __attribute__((ext_vector_type(8)));

// ---------------------------------------------------------------------------
// Kernel A: sigma[o,i] = exp(weight_psi[o,i])   (1M elements, 4 MB -> d_ws)
// ---------------------------------------------------------------------------
__global__ void sigma_exp_kernel(const float* __restrict__ psi,
                                 float* __restrict__ sigma) {
    int idx = (blockIdx.x * blockDim.x + threadIdx.x) * 4;
    v4f p = *(const v4f*)(psi + idx);
    v4f s;
    s.x = __expf(p.x);
    s.y = __expf(p.y);
    s.z = __expf(p.z);
    s.w = __expf(p.w);
    *(v4f*)(sigma + idx) = s;
}

// ---------------------------------------------------------------------------
// Kernel B: out[b,o] = sum_i mu[o,i]*x[b,i]  (f32 WMMA GEMM)
//                      + eps_b[b,o]*exp(bias_psi[o]) + bias_mu[o]
// One wave computes one 16x16 output tile via V_WMMA_F32_16X16X4_F32.
// A-matrix (x tile, 16x4): lanes 0-15 -> K=k0+0/1, lanes 16-31 -> K=k0+2/3,
// M = lane%16.  B-matrix (mu^T tile, 4x16): same K striping, N = lane%16.
// C/D: 8 VGPRs, element (M = r + (lane<16?0:8), N = lane%16).
// ---------------------------------------------------------------------------
__global__ void mu_gemm_bias_kernel(const float* __restrict__ x,
                                    const float* __restrict__ mu,
                                    const float* __restrict__ bias_mu,
                                    const float* __restrict__ bias_psi,
                                    const float* __restrict__ eps_b,
                                    float* __restrict__ out) {
    const int wave = (blockIdx.x * blockDim.x + threadIdx.x) >> 5;
    const int lane = threadIdx.x & 31;

    const int tb = wave >> 6;      // batch-tile index   (BS/16  = 16)
    const int to = wave & 63;      // out-tile index     (OUT/16 = 64)
    const int b0 = tb * 16;
    const int o0 = to * 16;

    const int m    = lane & 15;
    const int koff = (lane < 16) ? 0 : 2;

    const float* xrow = x  + (size_t)(b0 + m) * IN + koff;  // A: row M=m
    const float* mrow = mu + (size_t)(o0 + m) * IN + koff;  // B: col N=m

    v8f acc = {};
    for (int k0 = 0; k0 < IN; k0 += 4) {
        v2f a = *(const v2f*)(xrow + k0);
        v2f b = *(const v2f*)(mrow + k0);
        acc = __builtin_amdgcn_wmma_f32_16x16x4_f32(
            /*neg_a=*/false, a, /*neg_b=*/false, b,
            /*c_mod=*/(short)0, acc, /*reuse_a=*/false, /*reuse_b=*/false);
    }

    // Epilogue: add reparameterized bias, write the deterministic part of out.
    const int   col = o0 + m;
    const float eb  = __expf(bias_psi[col]);
    const float bm  = bias_mu[col];
    const int   rbase = b0 + ((lane < 16) ? 0 : 8);
#pragma unroll
    for (int r = 0; r < 8; ++r) {
        const int row = rbase + r;
        out[(size_t)row * OUT + col] = acc[r] + eps_b[(size_t)row * OUT + col] * eb + bm;
    }
}

// ---------------------------------------------------------------------------
// Kernel C: out[b,o] += sum_i eps_w[b,o,i] * sigma[o,i] * x[b,i]
// HBM-bound 1 GiB stream. One wave per (b,o) row; per iteration each lane
// loads a float4 -> 512 contiguous bytes per wave (coalesced b128 loads).
// eps_w is read once -> non-temporal hint; sigma/x stay hot in L2 (192 MB).
// ---------------------------------------------------------------------------
__global__ void eps_stream_kernel(const float* __restrict__ eps_w,
                                  const float* __restrict__ sigma,
                                  const float* __restrict__ x,
                                  float* __restrict__ out) {
    const int wave = (blockIdx.x * blockDim.x + threadIdx.x) >> 5;
    const int lane = threadIdx.x & 31;
    const int b = wave >> 10;      // / OUT
    const int o = wave & 1023;     // % OUT

    const float* erow = eps_w + (size_t)((size_t)b * OUT + o) * IN;
    const float* srow = sigma + (size_t)o * IN;
    const float* xrow = x     + (size_t)b * IN;

    float acc = 0.0f;
#pragma unroll
    for (int i0 = 0; i0 < IN; i0 += 128) {
        const int idx = i0 + lane * 4;
        v4f e  = __builtin_nontemporal_load((const v4f*)(erow + idx));
        v4f s  = *(const v4f*)(srow + idx);
        v4f xv = *(const v4f*)(xrow + idx);
        acc = fmaf(e.x * s.x, xv.x, acc);
        acc = fmaf(e.y * s.y, xv.y, acc);
        acc = fmaf(e.z * s.z, xv.z, acc);
        acc = fmaf(e.w * s.w, xv.w, acc);
    }

    // wave32 tree reduction
#pragma unroll
    for (int off = 16; off > 0; off >>= 1)
        acc += __shfl_down(acc, off, 32);

    if (lane == 0)
        out[(size_t)b * OUT + o] += acc;   // kernel B already wrote mu+bias part
}

// ---------------------------------------------------------------------------
extern "C" void kernel_launch(void* const* d_in, const int* in_sizes, int n_in,
                              void* d_out, int out_size, void* d_ws, size_t ws_size,
                              hipStream_t stream) {
    const float* x        = (const float*)d_in[0];
    const float* wmu      = (const float*)d_in[1];
    const float* wpsi     = (const float*)d_in[2];
    const float* bmu      = (const float*)d_in[3];
    const float* bpsi     = (const float*)d_in[4];
    const float* eps_w    = (const float*)d_in[5];
    const float* eps_b    = (const float*)d_in[6];
    float*       out      = (float*)d_out;
    float*       sigma    = (float*)d_ws;          // 4 MB scratch

    // A: sigma = exp(psi): 1M elems / (256 thr * 4 elems) = 1024 blocks
    sigma_exp_kernel<<<(OUT * IN / 4) / 256, 256, 0, stream>>>(wpsi, sigma);

    // B: mu GEMM + bias: (BS/16)*(OUT/16) = 1024 waves, 8 waves/block
    mu_gemm_bias_kernel<<<(BS / 16) * (OUT / 16) / 8, 256, 0, stream>>>(
        x, wmu, bmu, bpsi, eps_b, out);

    // C: 1 GiB eps stream: BS*OUT waves, 8 waves/block = 32768 blocks
    eps_stream_kernel<<<(BS * OUT) / 8, 256, 0, stream>>>(eps_w, sigma, x, out);
}